// Quantize_29437705847334
// MI455X (gfx1250) — compile-verified
//
#include <hip/hip_runtime.h>

// Soft scalar quantization (16 centers) for MI455X / gfx1250.
// Bandwidth-bound: 16MB read, 319MB written -> ~14.4us floor @ 23.3 TB/s.
// Scores 2*c*x - c^2 computed with V_WMMA_F32_16X16X4_F32 (exact f32),
// redistributed lane-local with ds_swizzle SWAPX16, NT b128 stores for one-hot.

typedef float v2f __attribute__((ext_vector_type(2)));
typedef float v4f __attribute__((ext_vector_type(4)));
typedef float v8f __attribute__((ext_vector_type(8)));

#define LOG2E 1.4426950408889634f

#if defined(__has_builtin)
#if __has_builtin(__builtin_amdgcn_wmma_f32_16x16x4_f32)
#define HAVE_WMMA_F32X4 1
#endif
#endif

__device__ __forceinline__ float swap16(float v) {
  // ds_swizzle group-of-32: xor=0x10, or=0, and=0x1f  (SWAPX16)
  return __int_as_float(__builtin_amdgcn_ds_swizzle(__float_as_int(v), 0x401f));
}

// Per-lane epilogue: s[j] = score of center j for this lane's element
// (any per-element additive shift of the true -(x-c)^2 is fine: softmax is
// shift-invariant and we subtract the row max exactly like jax.nn.softmax).
__device__ __forceinline__ void finish_element(
    long long e, const float* __restrict__ s, const float* __restrict__ cj,
    float* __restrict__ qbar, float* __restrict__ qhard,
    float* __restrict__ qsoft, float* __restrict__ onehot) {
  float m = s[0];
  float bc = cj[0];
  int idx = 0;
#pragma unroll
  for (int j = 1; j < 16; ++j) {  // first-wins, matches jnp.argmax
    if (s[j] > m) { m = s[j]; bc = cj[j]; idx = j; }
  }
  float sum = 0.0f, acc = 0.0f;
#pragma unroll
  for (int j = 0; j < 16; ++j) {
    float w = __builtin_amdgcn_exp2f((s[j] - m) * LOG2E);
    sum += w;
    acc = fmaf(w, cj[j], acc);
  }
  float qs = acc / sum;
  float qh = bc;
  float qb = qs + (qh - qs);  // reproduce straight-through blend rounding

  __builtin_nontemporal_store(qb, qbar + e);
  __builtin_nontemporal_store(qh, qhard + e);
  __builtin_nontemporal_store(qs, qsoft + e);
#pragma unroll
  for (int g = 0; g < 4; ++g) {
    v4f oh;
#pragma unroll
    for (int k = 0; k < 4; ++k) oh[k] = (4 * g + k == idx) ? 1.0f : 0.0f;
    __builtin_nontemporal_store(oh, (v4f*)(onehot + e * 16 + g * 4));  // 64B-aligned
  }
}

__global__ void __launch_bounds__(256) quantize16_kernel(
    const float* __restrict__ x, const float* __restrict__ centers,
    float* __restrict__ out, long long N) {
  float* qbar = out;
  float* qhard = out + N;
  float* qsoft = out + 2 * N;
  float* onehot = out + 3 * N;

  const int lane = (int)(threadIdx.x & 31);
  const bool lo = lane < 16;

  // centers broadcast (uniform -> s_load) for the per-lane epilogue
  float cj[16];
#pragma unroll
  for (int j = 0; j < 16; ++j) cj[j] = centers[j];

#if HAVE_WMMA_F32X4
  // A (16x4 f32, 2 VGPRs): lanes0-15 hold M=lane; VGPR0=K0, VGPR1=K1 (K2/K3 in hi lanes = 0).
  // Only K=0 used: A[m][0] = 2*c_m.
  const float c_lane = centers[lane & 15];
  v2f A;
  A[0] = lo ? (2.0f * c_lane) : 0.0f;
  A[1] = 0.0f;
  // C (16x16 f32 accum, 8 VGPRs): VGPR r = rows r (lanes0-15) and r+8 (lanes16-31).
  // C[m][n] = -c_m^2  (column-independent).
  v8f Cm;
#pragma unroll
  for (int r = 0; r < 8; ++r) {
    float cm = centers[(lo ? 0 : 8) + r];
    Cm[r] = -cm * cm;
  }
#endif

  const long long tiles = N >> 5;  // 32 elements per wave-tile
  const long long wave =
      (((long long)blockIdx.x * blockDim.x) + threadIdx.x) >> 5;
  const long long nwaves = ((long long)gridDim.x * blockDim.x) >> 5;

  for (long long t = wave; t < tiles; t += nwaves) {
    const long long base = t << 5;
    const float xv = __builtin_nontemporal_load(x + base + lane);
    float s[16];
#if HAVE_WMMA_F32X4
    // B (4x16 f32, 2 VGPRs): row K=0 in VGPR0 lanes0-15; other rows irrelevant (A K1..3 = 0).
    const float xs = swap16(xv);
    v2f B1; B1[0] = lo ? xv : 0.0f; B1[1] = 0.0f;  // columns = elements base+0..15
    v2f B2; B2[0] = lo ? xs : 0.0f; B2[1] = 0.0f;  // columns = elements base+16..31
    // D[m][n] = 2*c_m*x_n - c_m^2  (exact f32)
    v8f D1 = __builtin_amdgcn_wmma_f32_16x16x4_f32(false, A, false, B1,
                                                   (short)0, Cm, false, false);
    v8f D2 = __builtin_amdgcn_wmma_f32_16x16x4_f32(false, A, false, B2,
                                                   (short)0, Cm, false, false);
    // Redistribute: lane l < 16 owns element base+l (column l of D1);
    // lane l >= 16 owns element base+l (column l-16 of D2). Column n lives in
    // lanes n (rows 0-7) and n+16 (rows 8-15) -> one SWAPX16 per VGPR.
#pragma unroll
    for (int r = 0; r < 8; ++r) {
      float d1 = D1[r], d2 = D2[r];
      float d1s = swap16(d1);
      float d2s = swap16(d2);
      s[r] = lo ? d1 : d2s;       // center r
      s[r + 8] = lo ? d1s : d2;   // center r+8
    }
#else
    // Fallback: scalar scores (same shifted form)
#pragma unroll
    for (int j = 0; j < 16; ++j) s[j] = fmaf(2.0f * cj[j], xv, -cj[j] * cj[j]);
#endif
    finish_element(base + lane, s, cj, qbar, qhard, qsoft, onehot);
  }

  // Scalar tail for N % 32 != 0 (dead for the reference shape 4,194,304)
  const long long tail0 = tiles << 5;
  const long long gid = ((long long)blockIdx.x * blockDim.x) + threadIdx.x;
  const long long e = tail0 + gid;
  if (e < N) {
    float xv = x[e];
    float s[16];
#pragma unroll
    for (int j = 0; j < 16; ++j) {
      float d = xv - cj[j];
      s[j] = -d * d;
    }
    finish_element(e, s, cj, qbar, qhard, qsoft, onehot);
  }
}

extern "C" void kernel_launch(void* const* d_in, const int* in_sizes, int n_in,
                              void* d_out, int out_size, void* d_ws,
                              size_t ws_size, hipStream_t stream) {
  const float* x = (const float*)d_in[0];
  const float* centers = (const float*)d_in[1];
  float* out = (float*)d_out;
  const long long N = (long long)in_sizes[0];

  const int threads = 256;  // 8 wave32s per block
  long long tiles = N >> 5;
  long long blocksLL = (tiles + 63) / 64;  // ~8 tiles per wave at the cap
  if (blocksLL < 1) blocksLL = 1;
  if (blocksLL > 2048) blocksLL = 2048;    // grid-stride beyond this
  quantize16_kernel<<<(int)blocksLL, threads, 0, stream>>>(x, centers, out, N);
}